// ModelNew_58128087384725
// MI455X (gfx1250) — compile-verified
//
#include <hip/hip_runtime.h>
#include <hip/hip_bf16.h>
#include <math.h>

// ---------------------------------------------------------------------------
// Problem constants (from reference):
//   x: [8,32,16,32,32] f32, w: [32,64,3,3,3], conv_bias: [64], bias: [64]
//   ConvTranspose3d stride=2 pad=1 k=3 -> y[8,64,31,63,63]
//   out = 2*tanh(softmax_c(mean_D(y) + bias))  -> [8,64,1,63,63]
//
// Depth-mean folded analytically:
//   sum_{d'} y = conv2dT( A=[S, x0, x15] (96 ch), W2=[sum_kd w, -w[kd0], -w[kd2]] )
//   mean = conv_bias + Z/31
// ---------------------------------------------------------------------------

typedef __attribute__((ext_vector_type(2))) float v2f;
typedef __attribute__((ext_vector_type(8))) float v8f;

#define NB   8
#define NCI  32
#define ND   16
#define NHW  32
#define NCO  64
#define NQ   96          // 3 terms * 32 ci
#define HOUT 63
#define A_ELEMS  (NB*NHW*NHW*NQ)        // 786432 floats, layout [b][h][w][q]
#define W2_ELEMS (9*NQ*NCO)             // 55296 floats, paired layout (below)

// ---------------------------------------------------------------------------
// Kernel 1: depth reduction. One thread per (b,ci,h,w).
// A[b][h][w][q]:  q=ci -> sum_d x ; q=32+ci -> x[d=0] ; q=64+ci -> x[d=15]
// q is innermost so each WMMA lane loads its (q,q+1) pair with one b64.
// ---------------------------------------------------------------------------
__global__ __launch_bounds__(256) void k_reduce_d(const float* __restrict__ x,
                                                  float* __restrict__ A)
{
    int idx = blockIdx.x * blockDim.x + threadIdx.x;
    if (idx >= NB*NCI*NHW*NHW) return;
    int w  = idx & 31;
    int h  = (idx >> 5) & 31;
    int ci = (idx >> 10) & 31;
    int b  = idx >> 15;
    const float* xp = x + ((((size_t)b*NCI + ci)*ND)*NHW + h)*NHW + w; // d stride = 1024
    float s = 0.f, x0 = 0.f, x15 = 0.f;
#pragma unroll
    for (int d = 0; d < ND; ++d) {
        float t = xp[(size_t)d * (NHW*NHW)];
        s += t;
        if (d == 0)  x0  = t;
        if (d == 15) x15 = t;
    }
    float* ap = A + ((size_t)((b*NHW + h)*NHW) + w) * NQ;
    ap[ci]      = s;
    ap[32 + ci] = x0;
    ap[64 + ci] = x15;
}

// ---------------------------------------------------------------------------
// Kernel 2: weight folding into B-fragment-friendly paired layout:
//   W2[(((kh*3+kw)*48 + (q>>1))*64 + co)*2 + (q&1)]
// so a lane's two K values (q, q+1) are one contiguous b64 load.
// ---------------------------------------------------------------------------
__global__ __launch_bounds__(256) void k_fold_w(const float* __restrict__ w,
                                                float* __restrict__ W2)
{
    int idx = blockIdx.x * blockDim.x + threadIdx.x;
    if (idx >= NCI*NCO*9) return;
    int kw = idx % 3;
    int kh = (idx / 3) % 3;
    int co = (idx / 9) % NCO;
    int ci = idx / (9 * NCO);
    const float* wp = w + ((size_t)ci*NCO + co)*27 + kh*3 + kw;  // kd stride = 9
    float w0 = wp[0], w1 = wp[9], w2v = wp[18];
    size_t tap = kh*3 + kw;
    auto put = [&](int q, float v) {
        W2[((tap*48 + (q >> 1))*64 + co)*2 + (q & 1)] = v;
    };
    put(ci,      w0 + w1 + w2v);   // pairs with S
    put(32 + ci, -w0);             // boundary correction d'=-1
    put(64 + ci, -w2v);            // boundary correction d'=31
}

// ---------------------------------------------------------------------------
// Kernel 3: WMMA implicit-GEMM transpose-conv (2D, folded) + softmax + tanh.
// grid = (4 m-tiles, 63 h', 8 b), block = 128 (4 waves).
//   wave -> co tile (N=16), all waves share one M=16 pixel tile.
// m-tile 0: w'=0,2..30 | 1: w'=32..62 | 2: w'=1,3..31 | 3: w'=33..63(last pad)
// V_WMMA_F32_16X16X4_F32 fragment layout (wave32):
//   A 16x4: lanes0-15 M, VGPR0/1 = K{0,1}; lanes16-31 VGPR0/1 = K{2,3}
//   B 4x16: lanes = N, per-lane K pair {2*half, 2*half+1}
//   C 16x16: lane = N (co), VGPR r = pixel r + 8*half
// ---------------------------------------------------------------------------
__global__ __launch_bounds__(128) void k_gemm_softmax(
    const float* __restrict__ A, const float* __restrict__ W2,
    const float* __restrict__ conv_bias, const float* __restrict__ bias,
    float* __restrict__ out)
{
    const int mtile = blockIdx.x;
    const int hp    = blockIdx.y;      // h' in [0,62]
    const int b     = blockIdx.z;
    const int tid   = threadIdx.x;
    const int wave  = tid >> 5;
    const int lane  = tid & 31;
    const int half  = lane >> 4;
    const int l15   = lane & 15;
    const int co    = (wave << 4) + l15;

    __shared__ float zbuf[16 * NCO];   // [pixel][co]
    __shared__ float smax[16], ssum[16];

    const int w0p = (mtile == 0) ? 0 : (mtile == 1) ? 32 : (mtile == 2) ? 1 : 33;
    const int pw  = (mtile >= 2);
    const int ph  = hp & 1;

    v8f c = {};

    int khs[2], nkh, kws[2], nkw;
    if (ph == 0) { khs[0] = 1; nkh = 1; } else { khs[0] = 0; khs[1] = 2; nkh = 2; }
    if (pw == 0) { kws[0] = 1; nkw = 1; } else { kws[0] = 0; kws[1] = 2; nkw = 2; }

    for (int i = 0; i < nkh; ++i) {
        const int kh  = khs[i];
        const int hin = (hp + 1 - kh) >> 1;                   // always in [0,31]
        for (int j = 0; j < nkw; ++j) {
            const int kw    = kws[j];
            int win = ((w0p + 1 - kw) >> 1) + l15;            // contiguous per M-tile
            if (win > 31) win = 31;                           // pad pixel (discarded)
            const float* aptr = A + ((size_t)((b*NHW + hin)*NHW + win))*NQ + 2*half;
            const float* bptr = W2 + (((size_t)(kh*3 + kw)*48 + half)*64 + co)*2;
#pragma unroll
            for (int k = 0; k < 24; ++k) {                    // K = 96, 4 per WMMA
                v2f av = *(const v2f*)(aptr + k*4);           // q, q+1 contiguous
                v2f bv = *(const v2f*)(bptr + (size_t)k*256); // qpair += 2 -> +256
                c = __builtin_amdgcn_wmma_f32_16x16x4_f32(
                        false, av, false, bv, (short)0, c, false, false);
            }
        }
    }

    // Scale by 1/31, add biases, stage per-pixel channel vectors in LDS.
    const float cb = conv_bias[co] + bias[co];
#pragma unroll
    for (int r = 0; r < 8; ++r) {
        int m = r + 8*half;                                   // pixel index 0..15
        zbuf[m*NCO + co] = c[r] * (1.0f/31.0f) + cb;
    }
    __syncthreads();

    // Per-pixel softmax stats over 64 channels.
    if (tid < 16) {
        float mx = -3.0e38f;
        for (int q = 0; q < NCO; ++q) mx = fmaxf(mx, zbuf[tid*NCO + q]);
        float s = 0.f;
        for (int q = 0; q < NCO; ++q) s += __expf(zbuf[tid*NCO + q] - mx);
        smax[tid] = mx;
        ssum[tid] = s;
    }
    __syncthreads();

    // out[b][co][0][h'][w'] = 2*tanh(softmax)
#pragma unroll
    for (int i = 0; i < 8; ++i) {
        int idx = tid + i*128;            // 1024 = 16 pixels * 64 co
        int p   = idx >> 6;
        int q   = idx & 63;
        int wp  = w0p + 2*p;
        if (wp < HOUT) {
            float v = __expf(zbuf[p*NCO + q] - smax[p]) / ssum[p];
            out[(((size_t)b*NCO + q)*HOUT + hp)*HOUT + wp] = 2.0f * tanhf(v);
        }
    }
}

// ---------------------------------------------------------------------------
extern "C" void kernel_launch(void* const* d_in, const int* in_sizes, int n_in,
                              void* d_out, int out_size, void* d_ws, size_t ws_size,
                              hipStream_t stream)
{
    const float* x         = (const float*)d_in[0];
    const float* w         = (const float*)d_in[1];
    const float* conv_bias = (const float*)d_in[2];
    const float* bias      = (const float*)d_in[3];
    float* out = (float*)d_out;

    float* Abuf  = (float*)d_ws;             // 786432 floats
    float* W2buf = Abuf + A_ELEMS;           //  55296 floats  (~3.4 MB total)

    k_reduce_d<<<(NB*NCI*NHW*NHW + 255)/256, 256, 0, stream>>>(x, Abuf);
    k_fold_w  <<<(NCI*NCO*9      + 255)/256, 256, 0, stream>>>(w, W2buf);

    dim3 grid(4, HOUT, NB);                  // m-tile, h', batch
    k_gemm_softmax<<<grid, 128, 0, stream>>>(Abuf, W2buf, conv_bias, bias, out);
}